// MinibatchDiscrimination2d_39281770889264
// MI455X (gfx1250) — compile-verified
//
#include <hip/hip_runtime.h>
#include <hip/hip_bf16.h>

// Shapes from the reference
#define Bz      32
#define IN_FLT  128
#define Nn      64
#define Kk      4
#define T_SP    16
#define OUT_FLT 32
#define Ff      16
#define D_IN    768                  // 3*16*16
#define D_OUT   8192                 // 32*16*16
#define DOUTF   131072               // D_OUT * F

typedef __attribute__((ext_vector_type(2))) float v2f;
typedef __attribute__((ext_vector_type(8))) float v8f;

// ---------------------------------------------------------------------------
// Kernel A: strided conv (kernel==stride==4, VALID)  x(32,128,64,64) -> Xr(32,768)
// Xr[b, oc*256 + ty*16 + tx] = sum_{ic,kh,kw} x[b,ic,ty*4+kh,tx*4+kw]*w[oc,ic,kh,kw]
// ---------------------------------------------------------------------------
__global__ __launch_bounds__(256) void conv_k(const float* __restrict__ x,
                                              const float* __restrict__ w,
                                              float* __restrict__ Xr) {
    int o = blockIdx.x * 256 + threadIdx.x;          // 0 .. 24575
    int b  = o / D_IN;
    int rem = o - b * D_IN;
    int oc = rem >> 8;                                // /256
    int p  = rem & 255;
    int ty = p >> 4, tx = p & 15;
    const float* xb = x + (size_t)b * IN_FLT * Nn * Nn + (size_t)(ty * 4) * Nn + tx * 4;
    const float* wo = w + (size_t)oc * IN_FLT * 16;
    float s = 0.f;
    for (int ic = 0; ic < IN_FLT; ++ic) {
        const float* xc = xb + (size_t)ic * (Nn * Nn);
        const float* wc = wo + ic * 16;
#pragma unroll
        for (int kh = 0; kh < 4; ++kh)
#pragma unroll
            for (int kw = 0; kw < 4; ++kw)
                s = fmaf(xc[kh * Nn + kw], wc[kh * 4 + kw], s);
    }
    Xr[o] = s;
}

// ---------------------------------------------------------------------------
// Kernel B: M(32,131072) = Xr(32,768) @ T(768,131072) with V_WMMA_F32_16X16X4_F32.
// Each wave owns 16 columns of T's output and the FULL M=32 extent (two 16x16
// C tiles share each B fragment), so the 402 MB T matrix is streamed exactly
// once from HBM -> this kernel is pinned at the 23.3 TB/s roofline floor.
//
// fp32 fragment layouts (ISA 7.12.2):
//   A 16x4 : lane<16 holds {A[m=lane][k], A[m=lane][k+1]}, lane>=16 -> k+2,k+3
//   B 4x16 : lane<16 holds {B[k][n=lane], B[k+1][n]},      lane>=16 -> k+2,k+3
//   C 16x16: VGPR r = D[row r][n=lane]  (lane>=16 -> row r+8, n=lane-16)
// ---------------------------------------------------------------------------
__global__ __launch_bounds__(256) void gemm_wmma_k(const float* __restrict__ Xr,
                                                   const float* __restrict__ Tm,
                                                   float* __restrict__ M) {
    const int lane    = threadIdx.x & 31;
    const int wave    = threadIdx.x >> 5;
    const int tile    = blockIdx.x * 8 + wave;     // 0 .. 8191 N-tiles
    const int n0      = tile << 4;
    const int l15     = lane & 15;
    const int halfSel = lane >> 4;                 // 0: k,k+1   1: k+2,k+3
    const int kk_off  = halfSel << 1;

    v8f c0 = {};   // rows 0..15
    v8f c1 = {};   // rows 16..31

    const float* tcol = Tm + n0 + l15;             // column base for this lane

    for (int k = 0; k < D_IN; k += 4) {
        const int kk = k + kk_off;
        // A fragments (8-byte aligned vector loads; Xr is tiny & L2-resident)
        v2f a0 = *(const v2f*)(Xr + (size_t)l15 * D_IN + kk);
        v2f a1 = *(const v2f*)(Xr + (size_t)(l15 + 16) * D_IN + kk);
        // B fragment: two rows of T, 16 contiguous floats per 16-lane half
        v2f b;
        b.x = tcol[(size_t)kk * DOUTF];
        b.y = tcol[(size_t)(kk + 1) * DOUTF];
        // D = A*B + C   (8-arg probed form: neg_a, A, neg_b, B, c_mod, C, reuse_a, reuse_b)
        c0 = __builtin_amdgcn_wmma_f32_16x16x4_f32(false, a0, false, b, (short)0, c0, false, false);
        c1 = __builtin_amdgcn_wmma_f32_16x16x4_f32(false, a1, false, b, (short)0, c1, false, false);
    }

    const int colg  = n0 + l15;
    const int rbase = halfSel << 3;                // lane>=16 -> rows +8
#pragma unroll
    for (int r = 0; r < 8; ++r) {
        M[(size_t)(rbase + r) * DOUTF + colg]        = c0[r];
        M[(size_t)(rbase + r + 16) * DOUTF + colg]   = c1[r];
    }
}

// ---------------------------------------------------------------------------
// Kernel C: out[i,d] = sum_j exp(-sum_f |M[i,d,f]-M[j,d,f]|) - 1
// One block = 8 d-values; stage M[:, d0:d0+8, :] (16 KB) in LDS.
// thread = (i = tid&31, dl = tid>>5)
// ---------------------------------------------------------------------------
__global__ __launch_bounds__(256) void pairwise_k(const float* __restrict__ M,
                                                  float* __restrict__ outS) {
    __shared__ float Ml[8][32][Ff];                // [dlocal][sample][f]
    const int d0 = blockIdx.x * 8;
    for (int t = threadIdx.x; t < 8 * 32 * Ff; t += 256) {
        int f  = t & 15;
        int j  = (t >> 4) & 31;
        int dl = t >> 9;
        Ml[dl][j][f] = M[(size_t)j * DOUTF + (size_t)(d0 + dl) * Ff + f];
    }
    __syncthreads();
    const int i  = threadIdx.x & 31;
    const int dl = threadIdx.x >> 5;
    float s = 0.f;
    for (int j = 0; j < 32; ++j) {
        float acc = 0.f;
#pragma unroll
        for (int f = 0; f < Ff; ++f)
            acc += fabsf(Ml[dl][i][f] - Ml[dl][j][f]);
        s += __expf(-acc);
    }
    outS[(size_t)i * D_OUT + d0 + dl] = s - 1.0f;
}

// ---------------------------------------------------------------------------
// Kernel D: ConvTranspose (stride==kernel -> no overlap) + channel concat.
// out (32, 160, 64, 64): c<128 copy of x; c>=128:
//   out[b,128+oc, ty*4+kh, tx*4+kw] = sum_ic outS[b, ic*256+ty*16+tx] * wd[ic,oc,kh,kw]
// Each 256-thread block lands inside one (b,c) plane -> branch is uniform.
// ---------------------------------------------------------------------------
__global__ __launch_bounds__(256) void deconv_concat_k(const float* __restrict__ x,
                                                       const float* __restrict__ outS,
                                                       const float* __restrict__ wd,
                                                       float* __restrict__ out) {
    size_t idx = (size_t)blockIdx.x * 256 + threadIdx.x;
    int xy = (int)(idx & 4095);                    // position in 64x64 plane
    int c  = (int)((idx >> 12) % 160);
    int b  = (int)(idx / (160 * 4096));
    if (c < IN_FLT) {
        out[idx] = x[((size_t)b * IN_FLT + c) * 4096 + xy];
    } else {
        int oc = c - IN_FLT;
        int y  = xy >> 6, xx = xy & 63;
        int ty = y >> 2,  kh = y & 3;
        int tx = xx >> 2, kw = xx & 3;
        const float* ob = outS + (size_t)b * D_OUT + ty * 16 + tx;
        const float* wb = wd + oc * 16 + kh * 4 + kw;    // wd[ic,oc,kh,kw] = ic*512 + ...
        float s = 0.f;
#pragma unroll 8
        for (int ic = 0; ic < OUT_FLT; ++ic)
            s = fmaf(ob[ic * 256], wb[ic * 512], s);
        out[idx] = s;
    }
}

// ---------------------------------------------------------------------------
extern "C" void kernel_launch(void* const* d_in, const int* in_sizes, int n_in,
                              void* d_out, int out_size, void* d_ws, size_t ws_size,
                              hipStream_t stream) {
    const float* x        = (const float*)d_in[0];   // 32*128*64*64
    const float* w_conv   = (const float*)d_in[1];   // 3*128*4*4
    const float* Tm       = (const float*)d_in[2];   // 768*8192*16
    const float* w_deconv = (const float*)d_in[3];   // 32*32*4*4
    float* out = (float*)d_out;                      // 32*160*64*64

    // Workspace layout (~17.1 MB total)
    float* Xr   = (float*)d_ws;                      // 32*768
    float* M    = Xr + (size_t)Bz * D_IN;            // 32*131072
    float* outS = M + (size_t)Bz * DOUTF;            // 32*8192

    conv_k        <<<(Bz * D_IN) / 256, 256, 0, stream>>>(x, w_conv, Xr);
    gemm_wmma_k   <<<(DOUTF / 16) / 8, 256, 0, stream>>>(Xr, Tm, M);
    pairwise_k    <<<D_OUT / 8, 256, 0, stream>>>(M, outS);
    deconv_concat_k<<<(size_t)Bz * 160 * 4096 / 256, 256, 0, stream>>>(x, outS, w_deconv, out);
}